// CTCPredictionsLayer_55516747268900
// MI455X (gfx1250) — compile-verified
//
#include <hip/hip_runtime.h>
#include <hip/hip_bf16.h>

#define NEG_INF  (-1.0e30f)
#define KILLED   (-1.0e38f)
#define HASH_PRIME 1000003u
#define W 32          // beam width
#define NC 128        // num classes
#define TMAX 512
#define NCAND (W + W * NC)   // 4128 candidates
#define NEXT  (W * NC)       // 4096 extend candidates = 8 * NTH
#define NTH 512              // 16 wave32 waves

typedef float v2f __attribute__((ext_vector_type(2)));
typedef float v8f __attribute__((ext_vector_type(8)));
typedef unsigned int u32x4 __attribute__((ext_vector_type(4)));
typedef unsigned int u32x8 __attribute__((ext_vector_type(8)));

__device__ __forceinline__ float lse2(float a, float b) {
    float m = fmaxf(a, b);
    if (m < -5.0e29f) return NEG_INF;
    return m + __logf(__expf(a - m) + __expf(b - m));
}

__global__ __launch_bounds__(NTH)
void ctc_beam_kernel(const float* __restrict__ data,   // [T, B, C] log-probs
                     const int* __restrict__ dlen,     // [B]
                     int* __restrict__ out,            // [B, T]
                     int* __restrict__ bp_ws) {        // [B, T, W] backpointers
    __shared__ float s_lp[NC];
    __shared__ int   s_hash[W];
    __shared__ float s_pb[W], s_pnb[W], s_ptot[W];
    __shared__ int   s_lens[W], s_last[W];
    __shared__ float n_pb[W], n_pnb[W];
    __shared__ int   n_hash[W], n_meta[W];             // parent | cls<<5 | ext<<12
    __shared__ float cand_pb32[W];                     // stay-only pb
    __shared__ float s_tot32[W];                       // stay-only tot (for top-k)
    __shared__ float cand_pnb[NCAND];
    __shared__ int   cand_hash[NCAND];
    __shared__ float s_rv[16];
    __shared__ int   s_ri[16];
    __shared__ int   s_mcount;
    __shared__ int   s_mlist[256];

    const int b    = blockIdx.x;
    const int tid  = threadIdx.x;
    const int lane = tid & 31;
    const int wav  = tid >> 5;

    // ---- init beam state (matches reference init) ----
    if (tid < W) {
        s_hash[tid] = tid;                       // hashes0 = arange(W)
        s_pb[tid]   = (tid == 0) ? 0.0f : NEG_INF;
        s_pnb[tid]  = NEG_INF;
        s_lens[tid] = 0;
        s_last[tid] = -1;
    }
    out[b * TMAX + tid] = 0;                     // blank-fill output row (NTH == TMAX)
    int Lb = dlen[b];
    if (Lb > TMAX) Lb = TMAX;
    if (Lb < 0)    Lb = 0;
    const unsigned lds_lp = (unsigned)(size_t)(void*)&s_lp[0];  // LDS byte offset = addr[31:0]
    __syncthreads();

    for (int t = 0; t < Lb; ++t) {
        // ---- TDM: DMA this step's logp row (128 f32) into LDS ----
        const float* row = data + ((size_t)t * 64 + b) * NC;
        if (wav == 0) {
            const unsigned long long ga = (unsigned long long)(size_t)row;
            u32x4 g0;
            g0.x = 1u;                                           // count=1 (valid D#)
            g0.y = lds_lp;                                       // lds_addr
            g0.z = (unsigned)(ga & 0xFFFFFFFFull);               // global_addr[31:0]
            g0.w = (unsigned)((ga >> 32) & 0x1FFFFFFull)         // global_addr[56:32]
                   | 0x80000000u;                                // type=2 ("image")
            u32x8 g1;
            g1[0] = 0x00020000u;          // workgroup_mask=0, data_size=2 (4B)
            g1[1] = (128u << 16);         // tensor_dim0 = 128 (lo16)
            g1[2] = (1u << 16);           // tensor_dim0 hi = 0, tensor_dim1 = 1 (lo16)
            g1[3] = (128u << 16);         // tensor_dim1 hi = 0, tile_dim0 = 128
            g1[4] = 1u;                   // tile_dim1 = 1, tile_dim2 = 0
            g1[5] = 128u;                 // tensor_dim0_stride = 128
            g1[6] = 0u;
            g1[7] = 0u;
            asm volatile("tensor_load_to_lds %0, %1" :: "s"(g0), "s"(g1) : "memory");
            __builtin_amdgcn_s_wait_tensorcnt(0);
        }
        if (t + 1 < Lb && tid < 8) {
            const float* nrow = data + ((size_t)(t + 1) * 64 + b) * NC;
            __builtin_prefetch(nrow + tid * 16, 0, 1);   // global_prefetch_b8 (L2-resident)
        }
        if (tid == 0) s_mcount = 0;
        __syncthreads();

        // ---- stay candidates (same prefix) + ptot ----
        if (tid < W) {
            float pb = s_pb[tid], pnb = s_pnb[tid];
            float pt = lse2(pb, pnb);
            s_ptot[tid]    = pt;
            int last       = s_last[tid];
            cand_pb32[tid] = pt + s_lp[0];                      // blank emission
            cand_pnb[tid]  = (last >= 0) ? (pnb + s_lp[last]) : NEG_INF;
            cand_hash[tid] = s_hash[tid];
        }
        __syncthreads();

        // ---- extend candidates via WMMA outer-sum: D[p][c] = ptot[p]*1 + 1*lp[c] ----
        {
            const int i  = wav >> 3, j = wav & 7;    // 2 x 8 grid of 16x16 tiles
            const int p0 = i * 16, c0 = j * 16;
            const int hi = lane >> 4, l = lane & 15;
            v2f a, bb;
            // A 16x4 f32: lanes 0-15 hold K=0,1; lanes 16-31 hold K=2,3 (zeros)
            a.x  = hi ? 0.0f : s_ptot[p0 + l];
            a.y  = hi ? 0.0f : 1.0f;
            // B 4x16 f32: VGPR0 = rows K=0 / K=2, VGPR1 = rows K=1 / K=3
            bb.x = hi ? 0.0f : 1.0f;
            bb.y = hi ? 0.0f : s_lp[c0 + l];
            v8f acc = {0.f,0.f,0.f,0.f,0.f,0.f,0.f,0.f};
            acc = __builtin_amdgcn_wmma_f32_16x16x4_f32(
                      false, a, false, bb, (short)0, acc, false, false);
            const int   c   = c0 + l;                // D column = lane % 16
            const float lpc = s_lp[c];
            const bool  cz  = (c == 0);
            #pragma unroll
            for (int r = 0; r < 8; ++r) {
                const int p = p0 + (hi ? r + 8 : r); // D row layout per ISA table
                float v    = acc[r];                 // = ptot[p] + lp[c]
                float vrep = s_pb[p] + lpc;          // repeat label needs blank gap
                v = (s_last[p] == c) ? vrep : v;     // branchless -> v_cndmask
                v = cz ? NEG_INF : v;                // no blank extension
                const int idx  = W + p * NC + c;
                cand_pnb[idx]  = v;
                cand_hash[idx] = (int)((unsigned)s_hash[p] * HASH_PRIME + (unsigned)(c + 1));
            }
        }
        __syncthreads();

        // ---- find ext candidates whose hash equals an existing beam hash ----
        {
            int hq[W];
            #pragma unroll
            for (int k = 0; k < W; ++k) hq[k] = s_hash[k];   // register-cached
            #pragma unroll
            for (int ii = 0; ii < 8; ++ii) {
                const int idx = W + tid + ii * NTH;          // NEXT == 8*NTH exactly
                if (cand_pnb[idx] > -5.0e29f) {
                    const int h = cand_hash[idx];
                    int q = -1;
                    #pragma unroll
                    for (int k = W - 1; k >= 0; --k)         // first (smallest) match wins
                        if (hq[k] == h) q = k;
                    if (q >= 0) {
                        int slot = atomicAdd(&s_mcount, 1);  // ds_add_rtn
                        if (slot < 256) s_mlist[slot] = (q << 16) | idx;
                    }
                }
            }
        }
        __syncthreads();
        if (tid == 0) {
            // merge matched exts into their stay twins (logsumexp)
            int mc = s_mcount < 256 ? s_mcount : 256;
            for (int k = 0; k < mc; ++k) {
                int e = s_mlist[k];
                int q = e >> 16, idx = e & 0xFFFF;
                cand_pnb[q]   = lse2(cand_pnb[q], cand_pnb[idx]);
                cand_pnb[idx] = NEG_INF;
            }
            // stay-stay dedupe (equal-hash beams)
            for (int q2 = 1; q2 < W; ++q2)
                for (int q1 = 0; q1 < q2; ++q1)
                    if (cand_hash[q1] == cand_hash[q2]) {
                        cand_pb32[q1] = lse2(cand_pb32[q1], cand_pb32[q2]);
                        cand_pnb[q1]  = lse2(cand_pnb[q1],  cand_pnb[q2]);
                        cand_pb32[q2] = NEG_INF;
                        cand_pnb[q2]  = NEG_INF;
                        break;
                    }
            // stay totals (ext total == its pnb since ext pb = -inf)
            for (int q = 0; q < W; ++q)
                s_tot32[q] = lse2(cand_pb32[q], cand_pnb[q]);
        }
        __syncthreads();

        // ---- top-32 by tot: iterative block argmax (no transcendentals) ----
        for (int k = 0; k < W; ++k) {
            float bv = (tid < W) ? s_tot32[tid] : KILLED;
            int   bi = (tid < W) ? tid : NCAND;
            #pragma unroll
            for (int ii = 0; ii < 8; ++ii) {
                const int idx = W + tid + ii * NTH;
                const float tv = cand_pnb[idx];              // == tot for ext
                if (tv > bv) { bv = tv; bi = idx; }          // keep smallest idx on tie
            }
            for (int off = 16; off > 0; off >>= 1) {         // wave32 shuffle tree
                float ov = __shfl_down(bv, off);
                int   oi = __shfl_down(bi, off);
                if (ov > bv || (ov == bv && oi < bi)) { bv = ov; bi = oi; }
            }
            if (lane == 0) { s_rv[wav] = bv; s_ri[wav] = bi; }
            __syncthreads();
            if (tid == 0) {
                float wv = s_rv[0]; int wi = s_ri[0];
                for (int w2 = 1; w2 < 16; ++w2)
                    if (s_rv[w2] > wv || (s_rv[w2] == wv && s_ri[w2] < wi)) {
                        wv = s_rv[w2]; wi = s_ri[w2];
                    }
                const int sel    = wi;
                const int is_ext = sel >= W;
                const int parent = is_ext ? (sel - W) / NC : sel;
                const int cls    = is_ext ? (sel - W) % NC : 0;
                n_pb[k]   = is_ext ? NEG_INF : cand_pb32[sel];
                n_pnb[k]  = cand_pnb[sel];
                n_hash[k] = cand_hash[sel];
                n_meta[k] = parent | (cls << 5) | (is_ext << 12);
                if (is_ext) cand_pnb[sel] = KILLED;          // distinct picks:
                else        s_tot32[sel]  = KILLED;          // sentinel below all junk
            }
            __syncthreads();
        }

        // ---- commit new beams + backpointers ----
        int nl = 0, nlast = -1;
        if (tid < W) {
            const int meta   = n_meta[tid];
            const int parent = meta & 31;
            const int cls    = (meta >> 5) & 127;
            const int ext    = (meta >> 12) & 1;
            nl    = s_lens[parent] + ext;
            nlast = ext ? cls : s_last[parent];
            bp_ws[((size_t)b * TMAX + t) * W + tid] = meta;
        }
        __syncthreads();
        if (tid < W) {
            s_pb[tid]   = n_pb[tid];
            s_pnb[tid]  = n_pnb[tid];
            s_hash[tid] = n_hash[tid];
            s_lens[tid] = nl;
            s_last[tid] = nlast;
        }
        __syncthreads();
    }

    // ---- top-1 path: backtrack backpointers, emit labels ----
    if (tid == 0) {
        float bv = KILLED;
        int   bq = 0;
        for (int q = 0; q < W; ++q) {
            float tot = lse2(s_pb[q], s_pnb[q]);
            if (tot > bv) { bv = tot; bq = q; }
        }
        int  cur = bq;
        int  pos = s_lens[bq];
        int* orow = out + b * TMAX;
        for (int tt = Lb - 1; tt >= 0; --tt) {
            const int w2     = bp_ws[((size_t)b * TMAX + tt) * W + cur];
            const int parent = w2 & 31;
            const int cls    = (w2 >> 5) & 127;
            const int ext    = (w2 >> 12) & 1;
            if (ext) { --pos; if (pos >= 0 && pos < TMAX) orow[pos] = cls; }
            cur = parent;
        }
    }
}

extern "C" void kernel_launch(void* const* d_in, const int* in_sizes, int n_in,
                              void* d_out, int out_size, void* d_ws, size_t ws_size,
                              hipStream_t stream) {
    (void)in_sizes; (void)n_in; (void)out_size; (void)ws_size;
    const float* data = (const float*)d_in[0];   // [512, 64, 128] f32 log-probs
    const int*   dlen = (const int*)d_in[1];     // [64] int32
    int*         out  = (int*)d_out;             // [64, 512] int32 labels
    int*         bp   = (int*)d_ws;              // [64, 512, 32] backpointers (4 MB)

    ctc_beam_kernel<<<64, NTH, 0, stream>>>(data, dlen, out, bp);
}